// OneParamPairwiseEquilibrate_37134287242022
// MI455X (gfx1250) — compile-verified
//
#include <hip/hip_runtime.h>
#include <hip/hip_bf16.h>
#include <math.h>

typedef float v2f __attribute__((ext_vector_type(2)));
typedef float v8f __attribute__((ext_vector_type(8)));

#define CZf 64
#define DITHER_AMT 1.0e-4f

// ---------------- device helpers ----------------

__device__ __forceinline__ void quat_mul_d(const float* a, const float* b, float* o) {
  o[0] = a[0]*b[0] - a[1]*b[1] - a[2]*b[2] - a[3]*b[3];
  o[1] = a[0]*b[1] + a[1]*b[0] + a[2]*b[3] - a[3]*b[2];
  o[2] = a[0]*b[2] - a[1]*b[3] + a[2]*b[0] + a[3]*b[1];
  o[3] = a[0]*b[3] + a[1]*b[2] - a[2]*b[1] + a[3]*b[0];
}

__device__ __forceinline__ void quat_to_rot_d(const float* q, float* R) {
  float w=q[0], x=q[1], y=q[2], z=q[3];
  R[0]=1.0f-2.0f*(y*y+z*z); R[1]=2.0f*(x*y-w*z);       R[2]=2.0f*(x*z+w*y);
  R[3]=2.0f*(x*y+w*z);      R[4]=1.0f-2.0f*(x*x+z*z);  R[5]=2.0f*(y*z-w*x);
  R[6]=2.0f*(x*z-w*y);      R[7]=2.0f*(y*z+w*x);       R[8]=1.0f-2.0f*(x*x+y*y);
}

__device__ __forceinline__ void atomicMaxF(float* addr, float val) {
  int* ia = (int*)addr;
  int old = __float_as_int(*addr);
  while (__int_as_float(old) < val) {
    int assumed = old;
    old = atomicCAS(ia, assumed, __float_as_int(val));
    if (old == assumed) break;
  }
}

__device__ __forceinline__ unsigned hashu(unsigned x) {
  x ^= x >> 16; x *= 0x7feb352du;
  x ^= x >> 15; x *= 0x846ca68bu;
  x ^= x >> 16;
  return x;
}

__device__ __forceinline__ float hnorm(unsigned s) {
  unsigned a = hashu(s), b = hashu(s ^ 0x9E3779B9u);
  float u1 = a * (1.0f/4294967296.0f) + 1e-7f;
  float u2 = b * (1.0f/4294967296.0f);
  return sqrtf(-2.0f*logf(u1)) * cosf(6.28318530718f*u2);
}

// ---------------- kernels ----------------

// Pack W_update (64x6) and W_logits (64x1) into 64x16 row-major, zero padded.
__global__ void pack_w_k(const float* __restrict__ Wu, const float* __restrict__ Wl,
                         float* __restrict__ Wp) {
  int i = blockIdx.x * blockDim.x + threadIdx.x;
  if (i >= CZf*16) return;
  int row = i >> 4, col = i & 15;
  float v = 0.0f;
  if (col < 6)       v = Wu[row*6 + col];
  else if (col == 6) v = Wl[row];
  Wp[i] = v;
}

__global__ void init_nodes_k(float* __restrict__ segmax, float* __restrict__ denom, int nN) {
  int n = blockIdx.x * blockDim.x + threadIdx.x;
  if (n >= nN) return;
  segmax[n] = -3.0e38f;
  denom[n]  = 0.0f;
}

// Streaming GEMM: euf[e][0..6] = edge_features[e] @ Wpack + bias  (col7 pad)
// One wave per 16-edge tile, K=64 via 16x V_WMMA_F32_16X16X4_F32.
__global__ void gemm_wmma_k(const float* __restrict__ ef,
                            const float* __restrict__ Wp,
                            const float* __restrict__ bu,
                            const float* __restrict__ bl,
                            float* __restrict__ euf, int nE) {
  int gwave = (int)((blockIdx.x * blockDim.x + threadIdx.x) >> 5);
  int lane  = threadIdx.x & 31;
  int rowbase = gwave * 16;
  if (rowbase >= nE) return;
  int m  = lane & 15;      // A: row M  /  B,C: column N
  int hi = lane >> 4;      // K-half selector
  const float* arow = ef + (size_t)(rowbase + m) * CZf;
  v8f acc = {};
  #pragma unroll
  for (int k = 0; k < CZf; k += 4) {
    v2f a, b;
    a.x = arow[k + 2*hi + 0];
    a.y = arow[k + 2*hi + 1];
    b.x = Wp[(k + 2*hi + 0)*16 + m];
    b.y = Wp[(k + 2*hi + 1)*16 + m];
    acc = __builtin_amdgcn_wmma_f32_16x16x4_f32(
        /*neg_a=*/false, a, /*neg_b=*/false, b,
        /*c_mod=*/(short)0, acc, /*reuse_a=*/false, /*reuse_b=*/false);
  }
  float bias = (m < 6) ? bu[m] : ((m == 6) ? bl[0] : 0.0f);
  if (m < 8) {
    #pragma unroll
    for (int i = 0; i < 8; ++i) {
      int r = rowbase + i + 8*hi;          // C layout: VGPR i -> M = i + 8*hi
      euf[(size_t)r*8 + m] = acc[i] + bias;
    }
  }
}

// qn = normalize(quats); Rcur = quat_to_rot(qn); tcur = trans
__global__ void node_init_k(const float* __restrict__ quats, const float* __restrict__ trans,
                            float* __restrict__ qn, float* __restrict__ Rcur,
                            float* __restrict__ tcur, int nN) {
  int n = blockIdx.x * blockDim.x + threadIdx.x;
  if (n >= nN) return;
  float q[4];
  #pragma unroll
  for (int i = 0; i < 4; ++i) q[i] = quats[(size_t)n*4 + i];
  float inv = rsqrtf(q[0]*q[0]+q[1]*q[1]+q[2]*q[2]+q[3]*q[3] + 1e-30f);
  #pragma unroll
  for (int i = 0; i < 4; ++i) { q[i] *= inv; qn[(size_t)n*4 + i] = q[i]; }
  float R[9]; quat_to_rot_d(q, R);
  #pragma unroll
  for (int i = 0; i < 9; ++i) Rcur[(size_t)n*9 + i] = R[i];
  #pragma unroll
  for (int i = 0; i < 3; ++i) tcur[(size_t)n*3 + i] = trans[(size_t)n*3 + i];
}

__global__ void edge_max_k(const float* __restrict__ euf, const int* __restrict__ src,
                           float* __restrict__ segmax, int nE) {
  int e = blockIdx.x * blockDim.x + threadIdx.x;
  if (e >= nE) return;
  atomicMaxF(segmax + src[e], euf[(size_t)e*8 + 6]);
}

__global__ void seg_fix_k(float* __restrict__ segmax, int nN) {
  int n = blockIdx.x * blockDim.x + threadIdx.x;
  if (n >= nN) return;
  float v = segmax[n];
  if (!(v > -1.0e37f && v < 1.0e37f)) segmax[n] = 0.0f;
}

__global__ void edge_exp_k(float* __restrict__ euf, const int* __restrict__ src,
                           const float* __restrict__ segmax, float* __restrict__ denom, int nE) {
  int e = blockIdx.x * blockDim.x + threadIdx.x;
  if (e >= nE) return;
  int s = src[e];
  float ex = expf(euf[(size_t)e*8 + 6] - segmax[s]);
  euf[(size_t)e*8 + 7] = ex;
  atomicAdd(denom + s, ex);
}

// Precompute pair_R (9), pair_t (3), weight (1) per edge.
__global__ void edge_pair_k(const float* __restrict__ euf, const float* __restrict__ qn,
                            const float* __restrict__ Rn, const float* __restrict__ trans,
                            const int* __restrict__ src, const int* __restrict__ dst,
                            const float* __restrict__ denom,
                            float* __restrict__ pR, float* __restrict__ pT,
                            float* __restrict__ pw, int nE) {
  int e = blockIdx.x * blockDim.x + threadIdx.x;
  if (e >= nE) return;
  int d = dst[e], s = src[e];
  float qd[4], qs[4];
  #pragma unroll
  for (int i = 0; i < 4; ++i) { qd[i] = qn[(size_t)d*4+i]; qs[i] = qn[(size_t)s*4+i]; }
  float qdc[4] = { qd[0], -qd[1], -qd[2], -qd[3] };
  float qrel[4]; quat_mul_d(qdc, qs, qrel);
  // t_rel = Rn[dst]^T (trans[src] - trans[dst])
  float v0 = trans[(size_t)s*3+0] - trans[(size_t)d*3+0];
  float v1 = trans[(size_t)s*3+1] - trans[(size_t)d*3+1];
  float v2 = trans[(size_t)s*3+2] - trans[(size_t)d*3+2];
  const float* Rd = Rn + (size_t)d*9;
  float trel[3];
  #pragma unroll
  for (int i = 0; i < 3; ++i) trel[i] = Rd[0*3+i]*v0 + Rd[1*3+i]*v1 + Rd[2*3+i]*v2;
  float bcd[4]  = { 0.0f, euf[(size_t)e*8+0], euf[(size_t)e*8+1], euf[(size_t)e*8+2] };
  float tvec[3] = { euf[(size_t)e*8+3], euf[(size_t)e*8+4], euf[(size_t)e*8+5] };
  float tmp[4]; quat_mul_d(qrel, bcd, tmp);
  float pq[4];
  #pragma unroll
  for (int i = 0; i < 4; ++i) pq[i] = qrel[i] + tmp[i];
  float inv = rsqrtf(pq[0]*pq[0]+pq[1]*pq[1]+pq[2]*pq[2]+pq[3]*pq[3] + 1e-30f);
  #pragma unroll
  for (int i = 0; i < 4; ++i) pq[i] *= inv;
  float Rrel[9]; quat_to_rot_d(qrel, Rrel);
  float pt[3];
  #pragma unroll
  for (int i = 0; i < 3; ++i)
    pt[i] = trel[i] + Rrel[i*3+0]*tvec[0] + Rrel[i*3+1]*tvec[1] + Rrel[i*3+2]*tvec[2];
  float PRm[9]; quat_to_rot_d(pq, PRm);
  #pragma unroll
  for (int i = 0; i < 9; ++i) pR[(size_t)e*9 + i] = PRm[i];
  #pragma unroll
  for (int i = 0; i < 3; ++i) pT[(size_t)e*3 + i] = pt[i];
  pw[e] = euf[(size_t)e*8 + 7] / (denom[s] + 1e-16f);
}

__global__ void zero_sums_k(float* __restrict__ Rsum, float* __restrict__ tsum, int nN) {
  int i = blockIdx.x * blockDim.x + threadIdx.x;
  int n9 = nN * 9;
  if (i < n9) Rsum[i] = 0.0f;
  else if (i < n9 + nN*3) tsum[i - n9] = 0.0f;
}

__global__ void edge_accum_k(const float* __restrict__ Rcur, const float* __restrict__ tcur,
                             const float* __restrict__ pR, const float* __restrict__ pT,
                             const float* __restrict__ pw,
                             const int* __restrict__ src, const int* __restrict__ dst,
                             float* __restrict__ Rsum, float* __restrict__ tsum, int nE) {
  int e = blockIdx.x * blockDim.x + threadIdx.x;
  if (e >= nE) return;
  int d = dst[e], s = src[e];
  float Rd[9], Pr[9], Pt[3];
  #pragma unroll
  for (int i = 0; i < 9; ++i) { Rd[i] = Rcur[(size_t)d*9+i]; Pr[i] = pR[(size_t)e*9+i]; }
  #pragma unroll
  for (int i = 0; i < 3; ++i) Pt[i] = pT[(size_t)e*3+i];
  float w = pw[e];
  float* rs = Rsum + (size_t)s*9;
  #pragma unroll
  for (int i = 0; i < 3; ++i)
    #pragma unroll
    for (int j = 0; j < 3; ++j) {
      float v = Rd[i*3+0]*Pr[0*3+j] + Rd[i*3+1]*Pr[1*3+j] + Rd[i*3+2]*Pr[2*3+j];
      atomicAdd(rs + i*3 + j, v * w);
    }
  float* ts = tsum + (size_t)s*3;
  #pragma unroll
  for (int i = 0; i < 3; ++i) {
    float v = tcur[(size_t)d*3+i] + Rd[i*3+0]*Pt[0] + Rd[i*3+1]*Pt[1] + Rd[i*3+2]*Pt[2];
    atomicAdd(ts + i, v * w);
  }
}

// Dither + nearest-rotation projection (Newton polar iteration), then commit R,t.
__global__ void node_update_k(const float* __restrict__ Rsum, const float* __restrict__ tsum,
                              float* __restrict__ Rcur, float* __restrict__ tcur,
                              int nN, int step) {
  int n = blockIdx.x * blockDim.x + threadIdx.x;
  if (n >= nN) return;
  float X[9];
  #pragma unroll
  for (int i = 0; i < 9; ++i)
    X[i] = Rsum[(size_t)n*9+i] + DITHER_AMT * hnorm((unsigned)(n*9+i) ^ (0x85EBCA6Bu * (unsigned)(step+1)));
  for (int it = 0; it < 10; ++it) {
    float c00=X[4]*X[8]-X[5]*X[7], c01=X[5]*X[6]-X[3]*X[8], c02=X[3]*X[7]-X[4]*X[6];
    float c10=X[2]*X[7]-X[1]*X[8], c11=X[0]*X[8]-X[2]*X[6], c12=X[1]*X[6]-X[0]*X[7];
    float c20=X[1]*X[5]-X[2]*X[4], c21=X[2]*X[3]-X[0]*X[5], c22=X[0]*X[4]-X[1]*X[3];
    float det = X[0]*c00 + X[1]*c01 + X[2]*c02;
    float id  = 1.0f / (det + copysignf(1e-20f, det));
    X[0]=0.5f*(X[0]+c00*id); X[1]=0.5f*(X[1]+c01*id); X[2]=0.5f*(X[2]+c02*id);
    X[3]=0.5f*(X[3]+c10*id); X[4]=0.5f*(X[4]+c11*id); X[5]=0.5f*(X[5]+c12*id);
    X[6]=0.5f*(X[6]+c20*id); X[7]=0.5f*(X[7]+c21*id); X[8]=0.5f*(X[8]+c22*id);
  }
  float det = X[0]*(X[4]*X[8]-X[5]*X[7]) - X[1]*(X[3]*X[8]-X[5]*X[6]) + X[2]*(X[3]*X[7]-X[4]*X[6]);
  if (det < 0.0f) { X[2] = -X[2]; X[5] = -X[5]; X[8] = -X[8]; }
  #pragma unroll
  for (int i = 0; i < 9; ++i) Rcur[(size_t)n*9+i] = X[i];
  #pragma unroll
  for (int i = 0; i < 3; ++i) tcur[(size_t)n*3+i] = tsum[(size_t)n*3+i];
}

// R -> quat (argmax branch) -> rotvec -> quat ; concat with t.
__global__ void finalize_k(const float* __restrict__ Rcur, const float* __restrict__ tcur,
                           float* __restrict__ out, int nN) {
  int n = blockIdx.x * blockDim.x + threadIdx.x;
  if (n >= nN) return;
  const float* R = Rcur + (size_t)n*9;
  float m00=R[0],m01=R[1],m02=R[2],m10=R[3],m11=R[4],m12=R[5],m20=R[6],m21=R[7],m22=R[8];
  float cand[4][4] = {
    {1.0f+m00+m11+m22, m21-m12,            m02-m20,            m10-m01},
    {m21-m12,          1.0f+m00-m11-m22,   m01+m10,            m02+m20},
    {m02-m20,          m01+m10,            1.0f-m00+m11-m22,   m12+m21},
    {m10-m01,          m02+m20,            m12+m21,            1.0f-m00-m11+m22}};
  float sc[4] = { m00+m11+m22, m00, m11, m22 };
  int best = 0;
  for (int i = 1; i < 4; ++i) if (sc[i] > sc[best]) best = i;
  float q[4] = { cand[best][0], cand[best][1], cand[best][2], cand[best][3] };
  float inv = rsqrtf(q[0]*q[0]+q[1]*q[1]+q[2]*q[2]+q[3]*q[3] + 1e-30f);
  #pragma unroll
  for (int i = 0; i < 4; ++i) q[i] *= inv;
  float sgn = (q[0] < 0.0f) ? -1.0f : 1.0f;
  #pragma unroll
  for (int i = 0; i < 4; ++i) q[i] *= sgn;
  float vn = sqrtf(q[1]*q[1]+q[2]*q[2]+q[3]*q[3]);
  float theta = 2.0f * atan2f(vn, q[0]);
  float rv[3];
  if (vn < 1e-6f) { rv[0]=2.0f*q[1]; rv[1]=2.0f*q[2]; rv[2]=2.0f*q[3]; }
  else { float sc2 = theta / vn; rv[0]=sc2*q[1]; rv[1]=sc2*q[2]; rv[2]=sc2*q[3]; }
  float e0=rv[0]+1e-8f, e1=rv[1]+1e-8f, e2=rv[2]+1e-8f;
  float th = sqrtf(e0*e0+e1*e1+e2*e2);
  float w  = cosf(0.5f*th);
  float sh = sinf(0.5f*th) / th;
  float* o = out + (size_t)n*7;
  o[0] = w; o[1] = sh*rv[0]; o[2] = sh*rv[1]; o[3] = sh*rv[2];
  o[4] = tcur[(size_t)n*3+0]; o[5] = tcur[(size_t)n*3+1]; o[6] = tcur[(size_t)n*3+2];
}

// ---------------- host launch ----------------

extern "C" void kernel_launch(void* const* d_in, const int* in_sizes, int n_in,
                              void* d_out, int out_size, void* d_ws, size_t ws_size,
                              hipStream_t stream) {
  const float* ef    = (const float*)d_in[0];
  const float* quats = (const float*)d_in[1];
  const float* trans = (const float*)d_in[2];
  const float* Wu    = (const float*)d_in[3];
  const float* bu    = (const float*)d_in[4];
  const float* Wl    = (const float*)d_in[5];
  const float* bl    = (const float*)d_in[6];
  const int*   eidx  = (const int*)d_in[7];

  const int nE = in_sizes[0] / CZf;
  const int nN = in_sizes[1] / 4;
  const int* dst = eidx;         // edge_index[0]
  const int* src = eidx + nE;    // edge_index[1]

  float* w = (float*)d_ws;
  float* Wp     = w;  w += 1024;
  float* euf    = w;  w += (size_t)nE * 8;
  float* segmax = w;  w += nN;
  float* denom  = w;  w += nN;
  float* qn     = w;  w += (size_t)nN * 4;
  float* Rcur   = w;  w += (size_t)nN * 9;
  float* tcur   = w;  w += (size_t)nN * 3;
  float* Rsum   = w;  w += (size_t)nN * 9;
  float* tsum   = w;  w += (size_t)nN * 3;
  float* pR     = w;  w += (size_t)nE * 9;
  float* pT     = w;  w += (size_t)nE * 3;
  float* pw     = w;  w += nE;

  const int B = 256;
  dim3 blk(B);
  int gN  = (nN + B - 1) / B;
  int gE  = (nE + B - 1) / B;

  pack_w_k<<<(CZf*16 + B - 1)/B, blk, 0, stream>>>(Wu, Wl, Wp);
  init_nodes_k<<<gN, blk, 0, stream>>>(segmax, denom, nN);

  int ntiles = (nE + 15) / 16;                 // one wave per 16-edge tile
  long gemmThreads = (long)ntiles * 32;
  gemm_wmma_k<<<(unsigned)((gemmThreads + B - 1)/B), blk, 0, stream>>>(ef, Wp, bu, bl, euf, nE);

  node_init_k<<<gN, blk, 0, stream>>>(quats, trans, qn, Rcur, tcur, nN);
  edge_max_k<<<gE, blk, 0, stream>>>(euf, src, segmax, nE);
  seg_fix_k<<<gN, blk, 0, stream>>>(segmax, nN);
  edge_exp_k<<<gE, blk, 0, stream>>>(euf, src, segmax, denom, nE);
  edge_pair_k<<<gE, blk, 0, stream>>>(euf, qn, Rcur, trans, src, dst, denom, pR, pT, pw, nE);

  int gZ = (nN*12 + B - 1) / B;
  for (int step = 0; step < 3; ++step) {
    zero_sums_k<<<gZ, blk, 0, stream>>>(Rsum, tsum, nN);
    edge_accum_k<<<gE, blk, 0, stream>>>(Rcur, tcur, pR, pT, pw, src, dst, Rsum, tsum, nE);
    node_update_k<<<gN, blk, 0, stream>>>(Rsum, tsum, Rcur, tcur, nN, step);
  }

  finalize_k<<<gN, blk, 0, stream>>>(Rcur, tcur, (float*)d_out, nN);
}